// Episode_91164975825422
// MI455X (gfx1250) — compile-verified
//
#include <hip/hip_runtime.h>

// -------------------------------------------------------------------------
// Shapes (episode constants): B=65536, D=256, V=125726, T=4, A=50.
// Roofline: ~215 MB of required traffic -> ~9.5us at 23.3 TB/s. Memory bound.
// Strategy:
//   Kernel 1: one wave32 per 16 batch rows. Stream x / emb[head] / emb[tail]
//             with b128 loads, accumulate 8 f32 partial-sum vectors per lane,
//             then collapse per-row totals with V_WMMA_F32_16X16X4_F32
//             (B = ones matrix => D[m][n] = sum_k A[m][k]): the matrix unit
//             as a free wave-level reduction engine, full f32 precision.
//   Kernel 2: only the T-1 slice of logits/action_space is live. Stage a
//             256-row logits slab (50 KB) in LDS via the Tensor Data Mover
//             (tensor_load_to_lds + s_wait_tensorcnt) — one DMA descriptor
//             per block, no VGPR round-trip — then per-thread argmax +
//             softmax-at-argmax, one int4 gather from action_space[T-1].
// -------------------------------------------------------------------------

typedef __attribute__((ext_vector_type(2))) float v2f;
typedef __attribute__((ext_vector_type(4))) float v4f;
typedef __attribute__((ext_vector_type(8))) float v8f;
typedef __attribute__((ext_vector_type(4))) unsigned int u32x4;
typedef __attribute__((ext_vector_type(8))) int i32x8;
typedef __attribute__((ext_vector_type(4))) int i32x4;

#define HAS_WMMA_F32X4 __has_builtin(__builtin_amdgcn_wmma_f32_16x16x4_f32)
#define HAS_TDM        __has_builtin(__builtin_amdgcn_tensor_load_to_lds)

// Input per lane: a = two f32 partials. A-matrix 16x4 layout: lane m holds
// K=0,1 and lane m+16 holds K=2,3 for row M=m. With B[k][n] = 1.0 the WMMA
// produces D[m][n] = total row-m sum, replicated over all columns n.
// Result layout (C/D 16x16 f32): lanes 0-15 hold rows 0-7 in r[0..7],
// lanes 16-31 hold rows 8-15 in r[0..7].
__device__ __forceinline__ v8f rowsum16x4(v2f a) {
#if HAS_WMMA_F32X4
  v2f b = {1.0f, 1.0f};
  v8f c = {};
  return __builtin_amdgcn_wmma_f32_16x16x4_f32(
      /*neg_a=*/false, a, /*neg_b=*/false, b,
      /*c_mod=*/(short)0, c, /*reuse_a=*/false, /*reuse_b=*/false);
#else
  float t = a.x + a.y;
  t += __shfl_xor(t, 16, 32);          // lanes m and m+16 both hold row-m sum
  const int lane = (int)(threadIdx.x & 31u);
  const int base = (lane < 16) ? 0 : 8;
  v8f r;
#pragma unroll
  for (int j = 0; j < 8; ++j) r[j] = __shfl(t, base + j, 32);
  return r;
#endif
}

__global__ __launch_bounds__(256) void pearson_kernel(
    const float* __restrict__ ctx, const float* __restrict__ emb,
    const int* __restrict__ heads, const int* __restrict__ tails,
    float* __restrict__ out, int B) {
  const int tid  = (int)threadIdx.x;
  const int wave = tid >> 5;
  const int lane = tid & 31;
  const int mloc = lane & 15;          // row within the wave's 16-row tile
  const int half = lane >> 4;          // which 4-column slice this lane owns
  const int tileRow = (int)blockIdx.x * 128 + wave * 16;
  const int row = tileRow + mloc;

  const float* xrow = ctx + (size_t)row * 256;
  const int h = heads[row];
  const int t = tails[row];
  const float* y1row = emb + (size_t)h * 256;
  const float* y2row = emb + (size_t)t * 256;
  const int coff = half * 4;

  v4f sx = {}, sxx = {}, sy1 = {}, syy1 = {}, sxy1 = {},
      sy2 = {}, syy2 = {}, sxy2 = {};
#pragma unroll 4
  for (int k0 = 0; k0 < 256; k0 += 8) {
    v4f xv = *(const v4f*)(xrow  + k0 + coff);
    v4f y1 = *(const v4f*)(y1row + k0 + coff);
    v4f y2 = *(const v4f*)(y2row + k0 + coff);
    sx   += xv;        sxx  += xv * xv;
    sy1  += y1;        syy1 += y1 * y1;   sxy1 += xv * y1;
    sy2  += y2;        syy2 += y2 * y2;   sxy2 += xv * y2;
  }

  auto fold = [](v4f v) -> v2f {
    v2f r; r.x = v.x + v.y; r.y = v.z + v.w; return r;
  };
  // 8 WMMA reductions; compute is free, the kernel is bandwidth-bound.
  v8f Sx   = rowsum16x4(fold(sx));
  v8f Sxx  = rowsum16x4(fold(sxx));
  v8f Sy1  = rowsum16x4(fold(sy1));
  v8f Syy1 = rowsum16x4(fold(syy1));
  v8f Sxy1 = rowsum16x4(fold(sxy1));
  v8f Sy2  = rowsum16x4(fold(sy2));
  v8f Syy2 = rowsum16x4(fold(syy2));
  v8f Sxy2 = rowsum16x4(fold(sxy2));

  // Lanes 0 and 16 hold rows tileRow..+7 and tileRow+8..+15 respectively.
  if ((lane & 15) == 0) {
    const float invn = 1.0f / 256.0f;
    const int rbase = tileRow + half * 8;
#pragma unroll
    for (int j = 0; j < 8; ++j) {
      const float sxj = Sx[j], sxxj = Sxx[j];
      const float vx  = sxxj - sxj * sxj * invn;
      const float v1  = Syy1[j] - Sy1[j] * Sy1[j] * invn;
      const float v2  = Syy2[j] - Sy2[j] * Sy2[j] * invn;
      const float s1  = (Sxy1[j] - sxj * Sy1[j] * invn) * rsqrtf(vx * v1);
      const float s2  = (Sxy2[j] - sxj * Sy2[j] * invn) * rsqrtf(vx * v2);
      const int r = rbase + j;
      out[r]                    = s1;                       // out[0] = s1
      out[(size_t)B + r]        = s2;                       // out[1] = s2
      out[(size_t)3 * B + r]    = (s1 > s2) ? (float)heads[r]
                                            : (float)tails[r];  // start_entities
    }
  }
}

#define A_DIM 50
#define NPB   (256 * A_DIM)   // floats staged per block = 12800 (50 KB)

__global__ __launch_bounds__(256) void lastpolicy_kernel(
    const float* __restrict__ logits, const int* __restrict__ aspace,
    float* __restrict__ out, int B, int T) {
  __shared__ float sl[NPB];
  const int tid  = (int)threadIdx.x;
  const int row0 = (int)blockIdx.x * 256;
  const float* src = logits + ((size_t)(T - 1) * B + row0) * A_DIM;

#if HAS_TDM
  if (tid < 32) {  // wave 0 programs the Tensor Data Mover (EXEC-independent DMA)
    // LDS aperture keeps the byte offset in addr[31:0] (ISA 10.2 aperture map),
    // so truncating the generic pointer yields the D# lds_addr.
    const unsigned           lds_off = (unsigned)(uintptr_t)(void*)sl;
    const unsigned long long ga      = (unsigned long long)(uintptr_t)src;

    // D# group 0: count=1 user descriptor, lds_addr, 57-bit global_addr, type=2.
    u32x4 g0;
    g0[0] = 1u;
    g0[1] = lds_off;
    g0[2] = (unsigned)ga;
    g0[3] = (unsigned)((ga >> 32) & 0x1FFFFFFu) | (2u << 30);

    // D# group 1: data_size=4B; 1-D tile: tensor_dim0=tile_dim0=NPB,
    // tensor_dim1=tile_dim1=1, tensor_dim0_stride=NPB. No multicast/pad/iterate.
    i32x8 g1;
    g1[0] = (int)(2u << 16);                                   // data_size=4B
    g1[1] = (int)(((unsigned)NPB & 0xFFFFu) << 16);            // tensor_dim0[15:0]
    g1[2] = (int)(((unsigned)NPB >> 16) | (1u << 16));         // dim0[31:16] | tensor_dim1=1
    g1[3] = (int)(((unsigned)NPB & 0xFFFFu) << 16);            // tile_dim0=NPB
    g1[4] = 1;                                                 // tile_dim1=1, tile_dim2=0
    g1[5] = (int)NPB;                                          // tensor_dim0_stride
    g1[6] = 0;
    g1[7] = 0;

    i32x4 gz = {0, 0, 0, 0};   // groups 2/3 unused (<=2D tensor)
#if __has_include(<hip/amd_detail/amd_gfx1250_TDM.h>)
    i32x8 gz8 = {};
    __builtin_amdgcn_tensor_load_to_lds(g0, g1, gz, gz, gz8, 0);   // clang-23 arity
#else
    __builtin_amdgcn_tensor_load_to_lds(g0, g1, gz, gz, 0);        // ROCm 7.2 arity
#endif
#if __has_builtin(__builtin_amdgcn_s_wait_tensorcnt)
    __builtin_amdgcn_s_wait_tensorcnt(0);
#else
    asm volatile("s_wait_tensorcnt 0x0" ::: "memory");
#endif
  }
  __syncthreads();
#else
  for (int i = tid; i < NPB; i += 256) sl[i] = src[i];  // coalesced fallback
  __syncthreads();
#endif

  const int b = row0 + tid;
  const float* ml = &sl[tid * A_DIM];
  float mx = ml[0];
  int idx = 0;
#pragma unroll
  for (int a = 1; a < A_DIM; ++a) {
    const float v = ml[a];
    if (v > mx) { mx = v; idx = a; }   // first-max, matches jnp.argmax
  }
  float s = 0.0f;
#pragma unroll
  for (int a = 0; a < A_DIM; ++a) s += __expf(ml[a] - mx);

  // chosen = action_space[T-1, b, idx, :]; 16B-aligned int4 gather.
  const size_t base = (((size_t)(T - 1) * B + b) * A_DIM + idx) * 4;
  const int4 c4 = *(const int4*)(aspace + base);

  out[(size_t)2 * B + b] = 1.0f / s;       // last_scores = probs[argmax]
  out[(size_t)4 * B + b] = (float)c4.y;    // current_entities
  out[(size_t)5 * B + b] = (float)c4.z;    // current_timestamps
  out[(size_t)6 * B + b] = (float)c4.w;    // current_timestamps2
}

extern "C" void kernel_launch(void* const* d_in, const int* in_sizes, int n_in,
                              void* d_out, int out_size, void* d_ws, size_t ws_size,
                              hipStream_t stream) {
  const float* ctx    = (const float*)d_in[0];  // context_qa   (B, 256)
  const float* emb    = (const float*)d_in[1];  // embedding    (V, 256)
  const int*   heads  = (const int*)d_in[2];    // (B,)
  const int*   tails  = (const int*)d_in[3];    // (B,)
  const int*   aspace = (const int*)d_in[4];    // (T, B, 50, 4)
  const float* logits = (const float*)d_in[5];  // (T, B, 50)
  float* out = (float*)d_out;                   // 7*B floats, concat outputs

  const int B = in_sizes[2];
  const int T = (int)(in_sizes[5] / ((long long)B * A_DIM));

  pearson_kernel<<<B / 128, 256, 0, stream>>>(ctx, emb, heads, tails, out, B);
  lastpolicy_kernel<<<B / 256, 256, 0, stream>>>(logits, aspace, out, B, T);
}